// FK_layer_84963043049644
// MI455X (gfx1250) — compile-verified
//
#include <hip/hip_runtime.h>

typedef __attribute__((ext_vector_type(2))) float v2f;
typedef __attribute__((ext_vector_type(8))) float v8f;

#define NF 65536
#define NJ 22
#define COPY ((size_t)NF * 23 * 16)

// Rotation axes (raw, un-normalized, as in reference ROT_AXIS)
__constant__ float c_axis[NJ][3] = {
  {0,1,0},{1,0,0},{1,0,0},{1,0,0},
  {0,1,0},{1,0,0},{1,0,0},{1,0,0},
  {0,1,0},{1,0,0},{1,0,0},{1,0,0},
  {0.571f,0.f,0.821f},{0,1,0},{1,0,0},{1,0,0},
  {1,0,0},{0,0,-1},{1,0,0},{1,0,0},
  {0,1,0},{0,1,0}};
// M_SH translations = BODY_T[1:] * 1000
__constant__ float c_tsh[NJ][3] = {
  {33,0,95},{0,0,0},{0,0,45},{0,0,25},
  {11,0,99},{0,0,0},{0,0,45},{0,0,25},
  {-11,0,95},{0,0,0},{0,0,45},{0,0,25},
  {-34,0,21},{-1,0,67},{0,0,0},{0,0,45},
  {0,0,25},{34,-9,29},{0,0,0},{0,0,38},
  {0,0,0},{0,0,32}};
// joints whose parent is the root (locals index jl -> joint jl+1)
__constant__ int c_isbr[NJ] = {1,0,0,0, 1,0,0,0, 1,0,0,0, 1,0,0,0,0, 1,0,0,0,0};

__device__ __forceinline__ void store_joint(const float* S, float* out,
                                            size_t fbase, int jnt, int l) {
  // 64 floats (4 frames x 4x4, col-major in S) -> row-major global, 2 dwords/lane
  int tt = 2 * l;            // 0..62, even
  int ff = tt >> 4;          // frame in group
  int rc = tt & 15;          // r*4+c (row-major), even -> c in {0,2}
  int rr = rc >> 2;
  int cc = rc & 3;
  float v0 = S[ff * 16 + cc * 4 + rr];
  float v1 = S[ff * 16 + (cc + 1) * 4 + rr];
  size_t o = ((fbase + (size_t)ff) * 23 + (size_t)jnt) * 16 + (size_t)rc;
  *(float2*)(out + o)        = make_float2(v0, v1);
  *(float2*)(out + o + COPY) = make_float2(v0, v1);
}

__global__ __launch_bounds__(256) void fk_wmma_kernel(
    const float* __restrict__ base, const float* __restrict__ rotv,
    const float* __restrict__ roff, const float* __restrict__ emb,
    float* __restrict__ out)
{
  __shared__ float smem[8 * 128];          // 8 waves x (BASE 64 + CUR 64)
  const int l   = threadIdx.x & 31;
  const int wid = threadIdx.x >> 5;
  float* BASE = smem + wid * 128;
  float* CUR  = BASE + 64;

  const int grp = blockIdx.x * (blockDim.x >> 5) + wid;  // 4 frames / wave
  const size_t fbase = (size_t)grp * 4;

  const int m  = l & 15;
  const int fl = m >> 2;          // frame in group (same role for A and B)
  const int r  = m & 3;           // A: row in 4x4; B: column in 4x4
  const int hi = l >> 4;          // upper lane half -> K/rows {2,3}
  const int k0 = hi ? 2 : 0;
  const size_t fg = fbase + (size_t)fl;

  // ---------------- base transform (quaternion -> R, + root offset) --------
  {
    const float* bp = base + fg * 7;
    float t0 = bp[0], t1 = bp[1], t2 = bp[2];
    float qw = bp[3], qx = bp[4], qy = bp[5], qz = bp[6];
    float qn = sqrtf(qw*qw + qx*qx + qy*qy + qz*qz);
    qw /= qn; qx /= qn; qy /= qn; qz /= qn;
    float R00 = 1.f - 2.f*(qy*qy + qz*qz), R01 = 2.f*(qx*qy - qw*qz), R02 = 2.f*(qx*qz + qw*qy);
    float R10 = 2.f*(qx*qy + qw*qz), R11 = 1.f - 2.f*(qx*qx + qz*qz), R12 = 2.f*(qy*qz - qw*qx);
    float R20 = 2.f*(qx*qz - qw*qy), R21 = 2.f*(qy*qz + qw*qx), R22 = 1.f - 2.f*(qx*qx + qy*qy);
    const float* op = roff + fg * 3;
    float T0 = t0 + op[0], T1 = t1 + op[1], T2 = t2 + op[2];
    // this lane supplies baseM[r][k0], baseM[r][k0+1]
    float m0 = (r==0)?R00:(r==1)?R10:(r==2)?R20:0.f;
    float m1 = (r==0)?R01:(r==1)?R11:(r==2)?R21:0.f;
    float m2 = (r==0)?R02:(r==1)?R12:(r==2)?R22:0.f;
    float m3 = (r==0)?T0 :(r==1)?T1 :(r==2)?T2 :1.f;
    BASE[fl*16 + k0*4 + r]       = hi ? m2 : m0;
    BASE[fl*16 + (k0+1)*4 + r]   = hi ? m3 : m1;
  }
  __builtin_amdgcn_wave_barrier();

  store_joint(BASE, out, fbase, 0, l);   // joint 0 = baseM

  // ---------------- chain: G <- G @ L via V_WMMA_F32_16X16X4_F32 ----------
  for (int jl = 0; jl < NJ; ++jl) {
    // A operand: stacked running globals (or baseM at branch starts)
    const float* S = c_isbr[jl] ? BASE : CUR;
    v2f a, b;
    a.x = S[fl*16 + k0*4 + r];
    a.y = S[fl*16 + (k0+1)*4 + r];

    // B operand: this lane supplies L[k0][c], L[k0+1][c] with c = r
    float th = rotv[fg * NJ + jl] + emb[fg * NJ + jl];
    float sn, cs;
    sincosf(th, &sn, &cs);
    const int cc = r;
    float nx = c_axis[jl][0], ny = c_axis[jl][1], nz = c_axis[jl][2];
    float ncol = (cc==0) ? nx : (cc==1) ? ny : nz;
    float sk0  = (cc==0) ? 0.f : (cc==1) ? -nz :  ny;   // skew column cc
    float sk1  = (cc==0) ?  nz : (cc==1) ? 0.f : -nx;
    float sk2  = (cc==0) ? -ny : (cc==1) ?  nx : 0.f;
    float omc = 1.f - cs;
    float Rr0 = ((cc==0)?cs:0.f) + omc*nx*ncol + sn*sk0;
    float Rr1 = ((cc==1)?cs:0.f) + omc*ny*ncol + sn*sk1;
    float Rr2 = ((cc==2)?cs:0.f) + omc*nz*ncol + sn*sk2;
    float L0, L1, L2;
    if (jl == 17) {               // M_SH[17]: rotation about Y by ~45 deg
      const float A17 = 0.70738827f, B17 = 0.70682518f;
      L0 =  A17*Rr0 + B17*Rr2;
      L1 =  Rr1;
      L2 = -B17*Rr0 + A17*Rr2;
    } else { L0 = Rr0; L1 = Rr1; L2 = Rr2; }
    float rb0 = hi ? L2 : L0;          // rotation rows k0
    float rb1 = hi ? 0.f : L1;         // row 3 of rotation part = 0
    float tb0 = hi ? c_tsh[jl][2] : c_tsh[jl][0];  // column 3 = [t_sh; 1]
    float tb1 = hi ? 1.f : c_tsh[jl][1];
    bool isc3 = (cc == 3);
    b.x = isc3 ? tb0 : rb0;
    b.y = isc3 ? tb1 : rb1;

    v8f c = {};
    v8f d = __builtin_amdgcn_wmma_f32_16x16x4_f32(false, a, false, b,
                                                  (short)0, c, false, false);

    // scatter D's diagonal 4x4 blocks (new globals) into CUR, col-major
    int l4 = l >> 2;
    bool act = (l4==0) | (l4==1) | (l4==6) | (l4==7);
    if (act) {
      int ffw = ((l4 >> 1) & 2) | (l4 & 1);   // frame owning this column
      int ccw = l & 3;                        // column within frame
      bool up = (l4 & 1);                     // rows in d[4..7] vs d[0..3]
      float w0 = up ? d[4] : d[0];
      float w1 = up ? d[5] : d[1];
      float w2 = up ? d[6] : d[2];
      float w3 = up ? d[7] : d[3];
      *(float4*)(CUR + ffw*16 + ccw*4) = make_float4(w0, w1, w2, w3);
    }
    __builtin_amdgcn_wave_barrier();

    store_joint(CUR, out, fbase, jl + 1, l);
  }
}

extern "C" void kernel_launch(void* const* d_in, const int* in_sizes, int n_in,
                              void* d_out, int out_size, void* d_ws, size_t ws_size,
                              hipStream_t stream) {
  const float* base = (const float*)d_in[0];
  const float* rotv = (const float*)d_in[1];
  const float* roff = (const float*)d_in[2];
  const float* emb  = (const float*)d_in[3];
  float* out = (float*)d_out;
  (void)in_sizes; (void)n_in; (void)out_size; (void)d_ws; (void)ws_size;
  // 65536 frames / 4 frames per wave = 16384 waves; 8 waves per 256-thread block
  dim3 grid(NF / 4 / 8), block(256);
  hipLaunchKernelGGL(fk_wmma_kernel, grid, block, 0, stream,
                     base, rotv, roff, emb, out);
}